// MultiHeadAttention_79431125172249
// MI455X (gfx1250) — compile-verified
//
#include <hip/hip_runtime.h>
#include <math.h>

// Problem constants (from reference)
#define BB 32
#define SS 4096
#define EE 1024
#define HH 16
#define DD 64
#define SC 256            // S-rows per chunk block
#define NC (SS / SC)      // 16 chunks per batch

typedef float v2f __attribute__((ext_vector_type(2)));
typedef float v8f __attribute__((ext_vector_type(8)));

__device__ __forceinline__ v8f wmma_f32_16x16x4(v2f a, v2f b, v8f c) {
    // D = A(16x4 f32) * B(4x16 f32) + C(16x16 f32)
    return __builtin_amdgcn_wmma_f32_16x16x4_f32(false, a, false, b, (short)0, c,
                                                 false, false);
}

// ---------------- K1: q[b,:] = seq1[b]@Wq + bq ----------------
__global__ void __launch_bounds__(256)
qproj_kernel(const float* __restrict__ seq1, const float* __restrict__ Wq,
             const float* __restrict__ bq, float* __restrict__ q)
{
    int gid = blockIdx.x * 256 + threadIdx.x;   // B*1024
    int b = gid >> 10, j = gid & 1023;
    const float* s1 = seq1 + (size_t)b * EE;
    float acc = 0.f;
#pragma unroll 4
    for (int e = 0; e < EE; ++e) acc += s1[e] * Wq[(size_t)e * EE + j];
    q[gid] = acc + bq[j];
}

// ---------------- K2: U[b,h,e] = sum_d Wk[e, h*64+d] * q[b, h*64+d] --------
__global__ void __launch_bounds__(256)
uproj_kernel(const float* __restrict__ q, const float* __restrict__ Wk,
             float* __restrict__ U)
{
    int gid = blockIdx.x * 256 + threadIdx.x;   // B*H*E
    int b = gid >> 14;
    int rem = gid & 16383;
    int h = rem >> 10;
    int e = rem & 1023;
    const float* qv = q + (size_t)b * EE + h * DD;
    const float* wk = Wk + (size_t)e * EE + h * DD;
    float acc = 0.f;
#pragma unroll 8
    for (int d = 0; d < DD; ++d) acc += wk[d] * qv[d];
    U[gid] = acc;   // layout [b][h][e]
}

// ---------------- K3: fused score + split-softmax + y-partials -------------
__global__ void __launch_bounds__(256)
attn_chunk_kernel(const float* __restrict__ X, const int* __restrict__ mask,
                  const float* __restrict__ U,
                  float* __restrict__ Mpart, float* __restrict__ Lpart,
                  float* __restrict__ Ypart)
{
    const int blk = blockIdx.x;       // b*NC + c
    const int b   = blk / NC;
    const int s0  = (blk % NC) * SC;
    const int tid = threadIdx.x;
    const int w   = tid >> 5;         // wave 0..7
    const int l   = tid & 31;         // lane
    const int lo  = l & 15;
    const int hi  = l >> 4;           // 0/1

    __shared__ float Wlds[HH][SC + 4];   // weights, transposed [h][s], pad->bank-clean
    __shared__ float RedM[HH][32];
    __shared__ float RedL[HH][32];
    __shared__ float Mh[HH];

    const float* Xb = X + (size_t)b * SS * EE;
    const float* Ub = U + (size_t)b * HH * EE;

    // ---- Pass B: score[16s x 16h] per stripe; stripes sp = w and w+8 ----
    // A frag: lane l -> A[m=lo][k = k0 + 2*hi + v]  (v = vgpr 0/1)
    // B frag: lane l -> B[k = k0 + 2*hi + v][n=lo] = U[b][n][k...]
    v8f a0e = {}, a0o = {}, a1e = {}, a1o = {};
    const float* xr0 = Xb + (size_t)(s0 + w * 16 + lo) * EE + 2 * hi;
    const float* xr1 = Xb + (size_t)(s0 + (w + 8) * 16 + lo) * EE + 2 * hi;
    const float* ur  = Ub + (size_t)lo * EE + 2 * hi;
    for (int k0 = 0; k0 < EE; k0 += 8) {
        if ((k0 & 63) == 0) {
            __builtin_prefetch(xr0 + k0 + 512);
            __builtin_prefetch(xr1 + k0 + 512);
        }
        v2f bf0 = *(const v2f*)(ur + k0);
        v2f bf1 = *(const v2f*)(ur + k0 + 4);
        v2f x00 = *(const v2f*)(xr0 + k0);
        v2f x01 = *(const v2f*)(xr0 + k0 + 4);
        v2f x10 = *(const v2f*)(xr1 + k0);
        v2f x11 = *(const v2f*)(xr1 + k0 + 4);
        a0e = wmma_f32_16x16x4(x00, bf0, a0e);
        a1e = wmma_f32_16x16x4(x10, bf0, a1e);
        a0o = wmma_f32_16x16x4(x01, bf1, a0o);
        a1o = wmma_f32_16x16x4(x11, bf1, a1o);
    }
    v8f acc0 = a0e + a0o;
    v8f acc1 = a1e + a1o;

    // ---- masked split-softmax over this 256-row chunk ----
    // C/D layout: lane l holds head n = lo, rows m = 8*hi + v
    const float scl = 0.125f;                  // 1/sqrt(D)
    const float NEG = -__builtin_inff();
    float vals[2][8];
    int   okf[2][8];
    for (int t = 0; t < 2; ++t) {
        int sp  = w + 8 * t;
        int slb = sp * 16 + 8 * hi;            // s_local of v=0
        const int* mrow = mask + (size_t)b * SS + s0 + slb;
        v8f a = t ? acc1 : acc0;
        float mx = NEG;
        for (int v = 0; v < 8; ++v) {
            float sc = a[v] * scl;
            int ok = (mrow[v] != 0);
            vals[t][v] = sc;
            okf[t][v]  = ok;
            if (ok) mx = fmaxf(mx, sc);
        }
        RedM[lo][w * 4 + t * 2 + hi] = mx;
    }
    __syncthreads();
    if (tid < HH) {
        float m = NEG;
        for (int i = 0; i < 32; ++i) m = fmaxf(m, RedM[tid][i]);
        Mh[tid] = m;
    }
    __syncthreads();
    for (int t = 0; t < 2; ++t) {
        int sp  = w + 8 * t;
        int slb = sp * 16 + 8 * hi;
        float m = Mh[lo];
        float ssum = 0.f;
        for (int v = 0; v < 8; ++v) {
            float wt = okf[t][v] ? __expf(vals[t][v] - m) : 0.f;
            Wlds[lo][slb + v] = wt;
            ssum += wt;
        }
        RedL[lo][w * 4 + t * 2 + hi] = ssum;
    }
    __syncthreads();
    if (tid < HH) {
        float s = 0.f;
        for (int i = 0; i < 32; ++i) s += RedL[tid][i];
        Lpart[(size_t)blk * HH + tid] = s;
        Mpart[(size_t)blk * HH + tid] = Mh[tid];
    }

    // ---- Pass D: y_c[16h x E] = W[16h x 256s] @ X[256s x E] ----
    // Wave w owns E-columns [w*128, w*128+128): 8 n-tiles, 8 independent accs.
    v8f yacc[8] = {};
    const int ebase = w * 128 + lo;
    for (int k0 = 0; k0 < SC; k0 += 4) {
        v2f a = *(const v2f*)(&Wlds[lo][k0 + 2 * hi]);
        const float* xp = Xb + (size_t)(s0 + k0 + 2 * hi) * EE + ebase;
#pragma unroll
        for (int tt = 0; tt < 8; ++tt) {
            v2f bf;
            bf[0] = xp[tt * 16];
            bf[1] = xp[EE + tt * 16];
            yacc[tt] = wmma_f32_16x16x4(a, bf, yacc[tt]);
        }
    }
#pragma unroll
    for (int tt = 0; tt < 8; ++tt) {
        int e = w * 128 + tt * 16 + lo;
#pragma unroll
        for (int v = 0; v < 8; ++v) {
            int h = v + 8 * hi;          // D: lane-> n=lo(e), m=v+8*hi(h)
            Ypart[((size_t)blk * HH + h) * EE + e] = yacc[tt][v];
        }
    }
}

// ---------------- K4: combine split-softmax partials -> y[b,h,e] -----------
__global__ void __launch_bounds__(256)
combine_kernel(const float* __restrict__ Mpart, const float* __restrict__ Lpart,
               const float* __restrict__ Ypart, float* __restrict__ Y)
{
    const int bh = blockIdx.x;        // b*H + h
    const int b = bh / HH, h = bh % HH;
    __shared__ float sscale[NC];
    __shared__ float slinv;
    if (threadIdx.x == 0) {
        float m = -__builtin_inff();
        for (int c = 0; c < NC; ++c)
            m = fmaxf(m, Mpart[((size_t)b * NC + c) * HH + h]);
        float lsum = 0.f;
        for (int c = 0; c < NC; ++c) {
            float mc = Mpart[((size_t)b * NC + c) * HH + h];
            float sc = (mc < -1e30f) ? 0.f : __expf(mc - m);
            sscale[c] = sc;
            lsum += Lpart[((size_t)b * NC + c) * HH + h] * sc;
        }
        slinv = (lsum > 0.f) ? 1.f / lsum : 0.f;
    }
    __syncthreads();
    for (int e = threadIdx.x; e < EE; e += 256) {
        float acc = 0.f;
        for (int c = 0; c < NC; ++c)
            acc += Ypart[(((size_t)b * NC + c) * HH + h) * EE + e] * sscale[c];
        Y[((size_t)b * HH + h) * EE + e] = acc * slinv;
    }
}

// ---------------- K5: out[b, h*64+j] = y[b,h,:]@Wv[:, h*64+j] + bv ---------
__global__ void __launch_bounds__(256)
outproj_kernel(const float* __restrict__ Y, const float* __restrict__ Wv,
               const float* __restrict__ bv, float* __restrict__ out)
{
    int gid = blockIdx.x * 256 + threadIdx.x;   // B*E
    int b = gid >> 10, col = gid & 1023, h = col >> 6;
    const float* yr = Y + ((size_t)b * HH + h) * EE;
    float acc = 0.f;
#pragma unroll 4
    for (int e = 0; e < EE; ++e) acc += yr[e] * Wv[(size_t)e * EE + col];
    out[gid] = acc + bv[col];
}

extern "C" void kernel_launch(void* const* d_in, const int* in_sizes, int n_in,
                              void* d_out, int out_size, void* d_ws, size_t ws_size,
                              hipStream_t stream)
{
    (void)in_sizes; (void)n_in; (void)out_size; (void)ws_size;
    const float* seq1 = (const float*)d_in[0];
    const float* seq2 = (const float*)d_in[1];
    const int*   mask = (const int*)d_in[2];
    const float* Wq   = (const float*)d_in[3];
    const float* bq   = (const float*)d_in[4];
    const float* Wk   = (const float*)d_in[5];
    // d_in[6] = bk: per-head constant shift of scores -> cancels in softmax.
    const float* Wv   = (const float*)d_in[7];
    const float* bv   = (const float*)d_in[8];
    float* out = (float*)d_out;

    // workspace carve-up (floats): ~36.6 MB total
    float* ws    = (float*)d_ws;
    float* q     = ws;                                   // 32*1024
    float* U     = q + (size_t)BB * EE;                  // 32*16*1024
    float* Mpart = U + (size_t)BB * HH * EE;             // 512*16
    float* Lpart = Mpart + (size_t)BB * NC * HH;         // 512*16
    float* Ypart = Lpart + (size_t)BB * NC * HH;         // 512*16*1024
    float* Y     = Ypart + (size_t)BB * NC * HH * EE;    // 32*16*1024

    qproj_kernel  <<<(BB * EE) / 256, 256, 0, stream>>>(seq1, Wq, bq, q);
    uproj_kernel  <<<(BB * HH * EE) / 256, 256, 0, stream>>>(q, Wk, U);
    attn_chunk_kernel<<<BB * NC, 256, 0, stream>>>(seq2, mask, U, Mpart, Lpart, Ypart);
    combine_kernel<<<BB * HH, 256, 0, stream>>>(Mpart, Lpart, Ypart, Y);
    outproj_kernel<<<(BB * EE) / 256, 256, 0, stream>>>(Y, Wv, bv, out);
}